// KalmanGain_40020505264367
// MI455X (gfx1250) — compile-verified
//
#include <hip/hip_runtime.h>

typedef __attribute__((ext_vector_type(2))) float v2f;
typedef __attribute__((ext_vector_type(4))) float v4f;
typedef __attribute__((ext_vector_type(8))) float v8f;

#define NTHREADS 256   // 8 waves (wave32), 2 batches/wave -> 16 batches/block

// ---------------------------------------------------------------------------
// Branchless block-diagonal gather from LDS staging.
//   tile(i,j) = Mat_b0[i][j]      (i<8, j<8; matrix zero-padded to 8x8)
//             = Mat_b1[i-8][j-8]  (i>=8, j>=8)
//             = 0                 off-diagonal / outside nr x nc
// nr, nc are powers of two -> clamp via mask keeps the load in-bounds, the
// result is selected with v_cndmask (no exec-mask branching around ds_load).
// ---------------------------------------------------------------------------
__device__ __forceinline__ float bd1(const float* __restrict__ s, int b0, int b1,
                                     int stride, int nr, int nc, int i, int j) {
    int ii = i & 7, jj = j & 7;
    int ok = (int)((i >> 3) == (j >> 3)) & (int)(ii < nr) & (int)(jj < nc);
    int b  = (i >> 3) ? b1 : b0;
    float v = s[b * stride + (ii & (nr - 1)) * nc + (jj & (nc - 1))];
    return ok ? v : 0.0f;
}

// Paired gather of tile(i,j), tile(i,j+1) with j even: both elements live in
// the same row of the same 8x8 block and share one validity predicate
// (nc even), so this is a single aligned ds_load_b64 + cndmask.
__device__ __forceinline__ v2f bd2(const float* __restrict__ s, int b0, int b1,
                                   int stride, int nr, int nc, int i, int j) {
    int ii = i & 7, jj = j & 7;
    int ok = (int)((i >> 3) == (j >> 3)) & (int)(ii < nr) & (int)(jj < nc);
    int b  = (i >> 3) ? b1 : b0;
    const float* p = s + b * stride + (ii & (nr - 1)) * nc + (jj & (nc - 1));
    v2f v = *(const v2f*)p;   // 8-byte aligned: stride, ii*nc, jj all even
    v2f r;
    r.x = ok ? v.x : 0.0f;
    r.y = ok ? v.y : 0.0f;
    return r;
}

// ---------------------------------------------------------------------------
// 16x16x16 fp32 matmul via 4 chained V_WMMA_F32_16X16X4_F32.
// f32 operand layouts (ISA 7.12.2):
//   A 16x4 : lanes 0-15 M=lane, VGPR0/1 = K{0,1}; lanes 16-31 K{2,3}
//   B 4x16 : mirrored (one row striped across lanes within a VGPR)
//   C/D    : VGPR r = row (r + 8*half), lane%16 = col
// loadA2(i,k0) returns {A[i][k0], A[i][k0+1]}; loadB1(k,n) returns B[k][n].
// ---------------------------------------------------------------------------
template <typename FA, typename FB>
__device__ __forceinline__ v8f mm16(FA loadA2, FB loadB1, v8f acc, int half, int li) {
#pragma unroll
    for (int kt = 0; kt < 4; ++kt) {
        int k0 = 4 * kt + 2 * half;
        v2f a = loadA2(li, k0);
        v2f b;
        b.x = loadB1(k0, li);
        b.y = loadB1(k0 + 1, li);
        acc = __builtin_amdgcn_wmma_f32_16x16x4_f32(false, a, false, b,
                                                    (short)0, acc, false, false);
    }
    return acc;
}

// 4x4 inverse via adjugate (S is SPD and well-conditioned per the reference).
__device__ __forceinline__ void inv4x4(const float* __restrict__ m, float* __restrict__ o) {
    float i0  =  m[5]*m[10]*m[15] - m[5]*m[11]*m[14] - m[9]*m[6]*m[15]
               + m[9]*m[7]*m[14] + m[13]*m[6]*m[11] - m[13]*m[7]*m[10];
    float i4  = -m[4]*m[10]*m[15] + m[4]*m[11]*m[14] + m[8]*m[6]*m[15]
               - m[8]*m[7]*m[14] - m[12]*m[6]*m[11] + m[12]*m[7]*m[10];
    float i8  =  m[4]*m[9]*m[15] - m[4]*m[11]*m[13] - m[8]*m[5]*m[15]
               + m[8]*m[7]*m[13] + m[12]*m[5]*m[11] - m[12]*m[7]*m[9];
    float i12 = -m[4]*m[9]*m[14] + m[4]*m[10]*m[13] + m[8]*m[5]*m[14]
               - m[8]*m[6]*m[13] - m[12]*m[5]*m[10] + m[12]*m[6]*m[9];
    float i1  = -m[1]*m[10]*m[15] + m[1]*m[11]*m[14] + m[9]*m[2]*m[15]
               - m[9]*m[3]*m[14] - m[13]*m[2]*m[11] + m[13]*m[3]*m[10];
    float i5  =  m[0]*m[10]*m[15] - m[0]*m[11]*m[14] - m[8]*m[2]*m[15]
               + m[8]*m[3]*m[14] + m[12]*m[2]*m[11] - m[12]*m[3]*m[10];
    float i9  = -m[0]*m[9]*m[15] + m[0]*m[11]*m[13] + m[8]*m[1]*m[15]
               - m[8]*m[3]*m[13] - m[12]*m[1]*m[11] + m[12]*m[3]*m[9];
    float i13 =  m[0]*m[9]*m[14] - m[0]*m[10]*m[13] - m[8]*m[1]*m[14]
               + m[8]*m[2]*m[13] + m[12]*m[1]*m[10] - m[12]*m[2]*m[9];
    float i2  =  m[1]*m[6]*m[15] - m[1]*m[7]*m[14] - m[5]*m[2]*m[15]
               + m[5]*m[3]*m[14] + m[13]*m[2]*m[7] - m[13]*m[3]*m[6];
    float i6  = -m[0]*m[6]*m[15] + m[0]*m[7]*m[14] + m[4]*m[2]*m[15]
               - m[4]*m[3]*m[14] - m[12]*m[2]*m[7] + m[12]*m[3]*m[6];
    float i10 =  m[0]*m[5]*m[15] - m[0]*m[7]*m[13] - m[4]*m[1]*m[15]
               + m[4]*m[3]*m[13] + m[12]*m[1]*m[7] - m[12]*m[3]*m[5];
    float i14 = -m[0]*m[5]*m[14] + m[0]*m[6]*m[13] + m[4]*m[1]*m[14]
               - m[4]*m[2]*m[13] - m[12]*m[1]*m[6] + m[12]*m[2]*m[5];
    float i3  = -m[1]*m[6]*m[11] + m[1]*m[7]*m[10] + m[5]*m[2]*m[11]
               - m[5]*m[3]*m[10] - m[9]*m[2]*m[7] + m[9]*m[3]*m[6];
    float i7  =  m[0]*m[6]*m[11] - m[0]*m[7]*m[10] - m[4]*m[2]*m[11]
               + m[4]*m[3]*m[10] + m[8]*m[2]*m[7] - m[8]*m[3]*m[6];
    float i11 = -m[0]*m[5]*m[11] + m[0]*m[7]*m[9] + m[4]*m[1]*m[11]
               - m[4]*m[3]*m[9] - m[8]*m[1]*m[7] + m[8]*m[3]*m[5];
    float i15 =  m[0]*m[5]*m[10] - m[0]*m[6]*m[9] - m[4]*m[1]*m[10]
               + m[4]*m[2]*m[9] + m[8]*m[1]*m[6] - m[8]*m[2]*m[5];
    float det = m[0]*i0 + m[1]*i4 + m[2]*i8 + m[3]*i12;
    float r = 1.0f / det;
    o[0]=i0*r;  o[1]=i1*r;  o[2]=i2*r;  o[3]=i3*r;
    o[4]=i4*r;  o[5]=i5*r;  o[6]=i6*r;  o[7]=i7*r;
    o[8]=i8*r;  o[9]=i9*r;  o[10]=i10*r; o[11]=i11*r;
    o[12]=i12*r; o[13]=i13*r; o[14]=i14*r; o[15]=i15*r;
}

__global__ __launch_bounds__(NTHREADS)
void kalman_gain_wmma(const float* __restrict__ gF, const float* __restrict__ gH,
                      const float* __restrict__ gP, const float* __restrict__ gQ,
                      const float* __restrict__ gR, float* __restrict__ gKG) {
    __shared__ __attribute__((aligned(16))) float sF[16 * 64];
    __shared__ __attribute__((aligned(16))) float sP[16 * 64];
    __shared__ __attribute__((aligned(16))) float sQ[16 * 64];
    __shared__ __attribute__((aligned(16))) float sH[16 * 32];
    __shared__ __attribute__((aligned(16))) float sR[16 * 16];
    __shared__ __attribute__((aligned(16))) float sT[8 * 256];   // per-wave 16x16 tile
    __shared__ __attribute__((aligned(16))) float sS[16 * 16];   // S (4x4/batch)
    __shared__ __attribute__((aligned(16))) float sSi[16 * 16];  // S^-1
    __shared__ __attribute__((aligned(16))) float sKG[16 * 32];  // output staging

    const int tid  = threadIdx.x;
    const int lane = tid & 31;
    const int w    = tid >> 5;
    const int half = lane >> 4;
    const int li   = lane & 15;
    const int b0   = 2 * w;
    const int b1   = 2 * w + 1;
    const long gb  = (long)blockIdx.x * 16;  // first batch of this block

    // ---- stage inputs: exact-fit coalesced b128 copies (no loops)
    ((v4f*)sF)[tid] = ((const v4f*)(gF + gb * 64))[tid];            // 256 f4
    ((v4f*)sP)[tid] = ((const v4f*)(gP + gb * 64))[tid];            // 256 f4
    ((v4f*)sQ)[tid] = ((const v4f*)(gQ + gb * 64))[tid];            // 256 f4
    if (tid < 128) ((v4f*)sH)[tid] = ((const v4f*)(gH + gb * 32))[tid];
    if (tid < 64)  ((v4f*)sR)[tid] = ((const v4f*)(gR + gb * 16))[tid];
    __syncthreads();

    float* tw = &sT[w * 256];

    // ---- 1) T1 = Ftile @ Ptile
    v8f acc = {};
    acc = mm16([&](int i, int k0) { return bd2(sF, b0, b1, 64, 8, 8, i, k0); },
               [&](int k, int n)  { return bd1(sP, b0, b1, 64, 8, 8, k, n); },
               acc, half, li);
#pragma unroll
    for (int r = 0; r < 8; ++r) tw[(r + 8 * half) * 16 + li] = acc[r];
    __syncthreads();

    // ---- 2) Sigma = T1 @ F^T + Q
    v8f accS = {};
#pragma unroll
    for (int r = 0; r < 8; ++r) accS[r] = bd1(sQ, b0, b1, 64, 8, 8, r + 8 * half, li);
    accS = mm16([&](int i, int k0) { return *(const v2f*)(tw + i * 16 + k0); },
                [&](int k, int n)  { return bd1(sF, b0, b1, 64, 8, 8, n, k); },
                accS, half, li);
    __syncthreads();
#pragma unroll
    for (int r = 0; r < 8; ++r) tw[(r + 8 * half) * 16 + li] = accS[r];
    __syncthreads();

    // ---- 3) P12 = Sigma @ H^T   (H is 4x8, zero-padded)
    v8f accP = {};
    accP = mm16([&](int i, int k0) { return *(const v2f*)(tw + i * 16 + k0); },
                [&](int k, int n)  { return bd1(sH, b0, b1, 32, 4, 8, n, k); },
                accP, half, li);
    __syncthreads();
#pragma unroll
    for (int r = 0; r < 8; ++r) tw[(r + 8 * half) * 16 + li] = accP[r];
    __syncthreads();

    // ---- 4) S = H @ P12 + R
    v8f accX = {};
#pragma unroll
    for (int r = 0; r < 8; ++r) accX[r] = bd1(sR, b0, b1, 16, 4, 4, r + 8 * half, li);
    accX = mm16([&](int i, int k0) { return bd2(sH, b0, b1, 32, 4, 8, i, k0); },
                [&](int k, int n)  { return tw[k * 16 + n]; },
                accX, half, li);
    // extract 4x4 S per batch (batch0: rows/cols 0-3; batch1: rows/cols 8-11)
    if (half == 0 && li < 4) {
#pragma unroll
        for (int r = 0; r < 4; ++r) sS[b0 * 16 + r * 4 + li] = accX[r];
    }
    if (half == 1 && li >= 8 && li < 12) {
#pragma unroll
        for (int r = 0; r < 4; ++r) sS[b1 * 16 + r * 4 + (li - 8)] = accX[r];
    }
    __syncthreads();

    // ---- invert S (one thread per batch; 16 batches per block)
    if (tid < 16) inv4x4(&sS[tid * 16], &sSi[tid * 16]);
    __syncthreads();

    // ---- 5) KG = P12 @ Sinv
    v8f accK = {};
    accK = mm16([&](int i, int k0) { return *(const v2f*)(tw + i * 16 + k0); },
                [&](int k, int n)  { return bd1(sSi, b0, b1, 16, 4, 4, k, n); },
                accK, half, li);
    // KG per batch = rows 0-7, cols 0-3 of its diagonal block
    if (half == 0 && li < 4) {
#pragma unroll
        for (int r = 0; r < 8; ++r) sKG[b0 * 32 + r * 4 + li] = accK[r];
    }
    if (half == 1 && li >= 8 && li < 12) {
#pragma unroll
        for (int r = 0; r < 8; ++r) sKG[b1 * 32 + r * 4 + (li - 8)] = accK[r];
    }
    __syncthreads();

    // ---- coalesced nontemporal output store (write-once stream)
    if (tid < 128) {
        v4f v = ((const v4f*)sKG)[tid];
        __builtin_nontemporal_store(v, ((v4f*)(gKG + gb * 32)) + tid);
    }
}

extern "C" void kernel_launch(void* const* d_in, const int* in_sizes, int n_in,
                              void* d_out, int out_size, void* d_ws, size_t ws_size,
                              hipStream_t stream) {
    const float* F = (const float*)d_in[0];  // [B,8,8]
    const float* H = (const float*)d_in[1];  // [B,4,8]
    const float* P = (const float*)d_in[2];  // [B,8,8] Sigma_previous
    const float* Q = (const float*)d_in[3];  // [B,8,8]
    const float* R = (const float*)d_in[4];  // [B,4,4]
    float* out = (float*)d_out;              // [B,8,4]

    int B = in_sizes[0] / 64;
    int blocks = B / 16;  // 16 batches per 256-thread block (8 waves x 2 batches)
    kalman_gain_wmma<<<blocks, NTHREADS, 0, stream>>>(F, H, P, Q, R, out);
}